// SessionGraph3_66073776881730
// MI455X (gfx1250) — compile-verified
//
#include <hip/hip_runtime.h>
#include <hip/hip_bf16.h>
#include <math.h>

#define D 256
#define NUM_NODE 40000
#define N_ITEM 40960
#define N_TGT 2048
#define E_INT 327680
#define E_AGG 40960

typedef float v2f __attribute__((ext_vector_type(2)));
typedef float v8f __attribute__((ext_vector_type(8)));
typedef unsigned int u32x4 __attribute__((ext_vector_type(4)));
typedef int i32x4 __attribute__((ext_vector_type(4)));
typedef int i32x8 __attribute__((ext_vector_type(8)));

// ---------------- helpers ----------------

__device__ __forceinline__ float waveReduceSum(float v) {
#pragma unroll
  for (int o = 16; o > 0; o >>= 1) v += __shfl_xor(v, o);
  return v;
}

__device__ __forceinline__ float waveReduceMax(float v) {
#pragma unroll
  for (int o = 16; o > 0; o >>= 1) v = fmaxf(v, __shfl_xor(v, o));
  return v;
}

// order-preserving float atomic max (finite values; init with -inf)
__device__ __forceinline__ void atomicMaxFloat(float* addr, float val) {
  if (val >= 0.0f) atomicMax((int*)addr, __float_as_int(val));
  else             atomicMin((unsigned int*)addr, (unsigned int)__float_as_int(val));
}

// ---------------- elementwise utilities ----------------

__global__ void k_fill(float* __restrict__ p, float v, size_t n) {
  size_t i = (size_t)blockIdx.x * blockDim.x + threadIdx.x;
  size_t s = (size_t)gridDim.x * blockDim.x;
  for (; i < n; i += s) p[i] = v;
}

__global__ void k_scale2(const float* __restrict__ in, float* __restrict__ out, size_t n) {
  size_t i = (size_t)blockIdx.x * blockDim.x + threadIdx.x;
  size_t s = (size_t)gridDim.x * blockDim.x;
  for (; i < n; i += s) out[i] = 2.0f * in[i];
}

// out[row] = l2norm(src[idx[row]]) ; idx==nullptr -> identity gather. block = 256 = D.
__global__ void k_l2norm(const float* __restrict__ src, const int* __restrict__ idx,
                         float* __restrict__ out) {
  __shared__ float red[8];
  int row = blockIdx.x;
  int g = idx ? idx[row] : row;
  float v = src[(size_t)g * D + threadIdx.x];
  float ss = waveReduceSum(v * v);
  if ((threadIdx.x & 31) == 0) red[threadIdx.x >> 5] = ss;
  __syncthreads();
  float tot = 0.0f;
#pragma unroll
  for (int i = 0; i < 8; ++i) tot += red[i];
  out[(size_t)row * D + threadIdx.x] = v / fmaxf(sqrtf(tot), 1e-12f);
}

// ---------------- GAT layer kernels ----------------

// one wave per edge: e = sum_d h[src]*h[dst]*h_d[dis]*w_pi ; atomic segment max over dst
__global__ void k_gat_edge_e(const float* __restrict__ hv, const float* __restrict__ disEmb,
                             const int* __restrict__ dis, const int* __restrict__ src,
                             const int* __restrict__ dst, const float* __restrict__ wpi,
                             float* __restrict__ e_out, float* __restrict__ m_buf, int nE) {
  int edge = blockIdx.x * 8 + (threadIdx.x >> 5);
  if (edge >= nE) return;
  int lane = threadIdx.x & 31;
  int s = src[edge], t = dst[edge], dd = dis[edge];
  const float* hs = hv + (size_t)s * D;
  const float* ht = hv + (size_t)t * D;
  const float* hd = disEmb + (size_t)dd * D;
  float acc = 0.0f;
#pragma unroll
  for (int d = lane; d < D; d += 32) acc += hs[d] * ht[d] * hd[d] * wpi[d];
  acc = waveReduceSum(acc);
  if (lane == 0) {
    e_out[edge] = acc;
    atomicMaxFloat(&m_buf[t], acc);
  }
}

__global__ void k_gat_expsum(const float* __restrict__ e, const int* __restrict__ dst,
                             const float* __restrict__ m, float* __restrict__ ex,
                             float* __restrict__ denom, int nE) {
  int i = blockIdx.x * 256 + threadIdx.x;
  if (i < nE) {
    int t = dst[i];
    float v = expf(e[i] - m[t]);
    ex[i] = v;
    atomicAdd(&denom[t], v);
  }
}

// one block (256 = D threads) per edge: hacc[dst] += hv[src] * (ex/denom[dst])
__global__ void k_gat_scatter(const float* __restrict__ hv, const float* __restrict__ ex,
                              const float* __restrict__ denom, const int* __restrict__ src,
                              const int* __restrict__ dst, float* __restrict__ hacc) {
  int edge = blockIdx.x;
  int s = src[edge], t = dst[edge];
  float a = ex[edge] / denom[t];
  atomicAdd(&hacc[(size_t)t * D + threadIdx.x], hv[(size_t)s * D + threadIdx.x] * a);
}

// ---------------- PosAggregator (fp32 WMMA GEMM fused with tanh/coef/scatter) ----------------
// 16 edges per block, 128 threads (4 waves). e = tanh(X @ w_q), X = [h[src]|pos_emb[pid]] (16x512)
// coef[r] = sum_j e[r,j]*tgt_emb[tid[dst[r]]][j] ; sr[dst[r]] += X[r,0:256]*coef[r]
__global__ void __launch_bounds__(128)
k_posagg(const float* __restrict__ h, const float* __restrict__ posEmb,
         const float* __restrict__ tgtEmb, const int* __restrict__ pid,
         const int* __restrict__ tidv, const int* __restrict__ srcA,
         const int* __restrict__ dstA, const float* __restrict__ wq,
         float* __restrict__ sr) {
  __shared__ float X[16][516];   // stride 516: bank 4*row + k, conflict-free; 16B aligned rows
  __shared__ float lcoef[16];
  __shared__ int   tIdxS[16], dstS[16];

  int e0 = blockIdx.x * 16;
  int t = threadIdx.x;
  if (t < 16) {
    lcoef[t] = 0.0f;
    int dn = dstA[e0 + t];
    dstS[t] = dn;
    tIdxS[t] = tidv[dn];
  }
  // stage X = [h[src] | posEmb[pid]] with float4 (both halves 16B aligned)
  for (int i = t; i < 16 * 64; i += 128) {
    int r = i >> 6, c4 = (i & 63) * 4;
    *(float4*)&X[r][c4] =
        *(const float4*)&h[(size_t)srcA[e0 + r] * D + c4];
    *(float4*)&X[r][256 + c4] =
        *(const float4*)&posEmb[(size_t)pid[e0 + r] * D + c4];
  }
  __syncthreads();

  int wave = t >> 5, lane = t & 31;
  int half = lane >> 4, idx = lane & 15;

  for (int q = 0; q < 4; ++q) {
    int n0 = (wave * 4 + q) * 16;
    v8f c = {};
#pragma unroll 4
    for (int k = 0; k < 2 * D; k += 4) {
      int kk = k + 2 * half;
      v2f a = *(const v2f*)&X[idx][kk];
      v2f b;
      b.x = wq[(size_t)kk * D + n0 + idx];
      b.y = wq[(size_t)(kk + 1) * D + n0 + idx];
      c = __builtin_amdgcn_wmma_f32_16x16x4_f32(false, a, false, b, (short)0, c,
                                                false, false);
    }
#pragma unroll
    for (int r = 0; r < 8; ++r) {
      int m = r + half * 8;                               // C layout: M = r + 8*(lane>=16)
      float ev = tanhf(c[r]);
      float contrib = ev * tgtEmb[(size_t)tIdxS[m] * D + n0 + idx];  // N = n0 + (lane&15)
      atomicAdd(&lcoef[m], contrib);
    }
  }
  __syncthreads();

  for (int r = 0; r < 16; ++r) {
    float cf = lcoef[r];
    int dn = dstS[r];
    for (int c = t; c < D; c += 128)
      atomicAdd(&sr[(size_t)dn * D + c], X[r][c] * cf);
  }
}

// ---------------- scoring GEMM: out = 12 * (sr_n @ b^T), fp32 WMMA + TDM staging ----------------
// grid (NUM_NODE/64, N_TGT/32), block 128 (4 waves).
// A tile (32x256 f32) is staged global->LDS by the Tensor Data Mover with hardware padding
// (256 DWORD rows + 4 DWORD pad -> LDS row stride 260 = conflict-free banks 4*row+k).
// Each wave owns one 16-wide N-subtile and TWO 16-row M-subtiles sharing each B fragment.
__global__ void __launch_bounds__(128)
k_logits(const float* __restrict__ A,   // [N_TGT, D] l2-normalized sr
         const float* __restrict__ B,   // [NUM_NODE, D] l2-normalized emb
         float* __restrict__ out) {     // [N_TGT, NUM_NODE]
  __shared__ float As[32][D + 4];       // stride 260: bank 4*row + k, conflict-free
  int m0 = blockIdx.y * 32;
  int nblk = blockIdx.x * 64;
  int t = threadIdx.x;

  int wave = t >> 5, lane = t & 31;
  int half = lane >> 4, idx = lane & 15;
  int n0 = nblk + wave * 16;
  const float* Brow = B + (size_t)(n0 + idx) * D;
  // pull this lane's B row toward the WGP while the TDM stages the A tile
  __builtin_prefetch(Brow, 0, 3);
  __builtin_prefetch(Brow + 64, 0, 3);
  __builtin_prefetch(Brow + 128, 0, 3);
  __builtin_prefetch(Brow + 192, 0, 3);

  if (wave == 0) {
    // ---- Tensor DMA descriptor (ISA 8.3-8.6): 2D tile, f32, padded LDS dest ----
    unsigned lds_addr = (unsigned)(size_t)&As[0][0];
    unsigned long long gaddr = (unsigned long long)(size_t)(A + (size_t)m0 * D);
    u32x4 g0;
    g0.x = 1u;                                  // count=1, is_restore=0, gather off
    g0.y = lds_addr;                            // lds_addr [63:32]
    g0.z = (unsigned)(gaddr & 0xFFFFFFFFull);   // global_addr [95:64]
    g0.w = (unsigned)(gaddr >> 32) | (2u << 30);// global_addr [120:96] | type=2 [127:126]
    i32x8 g1;
    g1[0] = (int)((2u << 16)      // data_size = 4B
                | (1u << 20)      // pad_enable
                | (7u << 22)      // pad_interval: 256 DWORDs
                | (3u << 25));    // pad_amount: 4 DWORDs
    g1[1] = (int)(256u << 16);    // tensor_dim0 lo16 at [63:48]
    g1[2] = (int)(32u << 16);     // tensor_dim0 hi16 = 0 | tensor_dim1 lo16 = 32
    g1[3] = (int)(256u << 16);    // tensor_dim1 hi16 = 0 | tile_dim0 = 256
    g1[4] = 32;                   // tile_dim1 = 32, tile_dim2 = 0
    g1[5] = 256;                  // tensor_dim0_stride lo32
    g1[6] = 0;                    // stride hi | tensor_dim1_stride lo
    g1[7] = 0;
    i32x4 gz4 = {0, 0, 0, 0};
    i32x8 gz8 = {0, 0, 0, 0, 0, 0, 0, 0};
    __builtin_amdgcn_tensor_load_to_lds(g0, g1, gz4, gz4, gz8, 0);
    __builtin_amdgcn_s_wait_tensorcnt(0);
  }
  __syncthreads();

  v8f c0 = {}, c1 = {};
#pragma unroll 4
  for (int k = 0; k < D; k += 4) {
    int kk = k + 2 * half;                  // lanes 0-15: K=k,k+1 ; lanes 16-31: K=k+2,k+3
    v2f b  = *(const v2f*)&Brow[kk];        // B^T[kk..kk+1, n0+idx]
    v2f a0 = *(const v2f*)&As[idx][kk];     // A[m0+idx,      kk..kk+1]
    v2f a1 = *(const v2f*)&As[16 + idx][kk];// A[m0+16+idx,   kk..kk+1]
    c0 = __builtin_amdgcn_wmma_f32_16x16x4_f32(false, a0, false, b, (short)0, c0,
                                               false, false);
    c1 = __builtin_amdgcn_wmma_f32_16x16x4_f32(false, a1, false, b, (short)0, c1,
                                               false, false);
  }
#pragma unroll
  for (int r = 0; r < 8; ++r) {
    int m = m0 + r + half * 8;              // C layout: M = r + 8*(lane>=16)
    out[(size_t)m * NUM_NODE + n0 + idx]        = 12.0f * c0[r];
    out[(size_t)(m + 16) * NUM_NODE + n0 + idx] = 12.0f * c1[r];
  }
}

// ---------------- log_softmax over rows of 40000 ----------------

__global__ void k_rowlse(const float* __restrict__ logits, float* __restrict__ lse) {
  __shared__ float redm[8], reds[8];
  int row = blockIdx.x;
  const float* p = logits + (size_t)row * NUM_NODE;
  float mx = -INFINITY;
  for (int i = threadIdx.x; i < NUM_NODE; i += 256) mx = fmaxf(mx, p[i]);
  mx = waveReduceMax(mx);
  if ((threadIdx.x & 31) == 0) redm[threadIdx.x >> 5] = mx;
  __syncthreads();
  mx = redm[0];
#pragma unroll
  for (int i = 1; i < 8; ++i) mx = fmaxf(mx, redm[i]);
  float s = 0.0f;
  for (int i = threadIdx.x; i < NUM_NODE; i += 256) s += expf(p[i] - mx);
  s = waveReduceSum(s);
  if ((threadIdx.x & 31) == 0) reds[threadIdx.x >> 5] = s;
  __syncthreads();
  if (threadIdx.x == 0) {
    float tot = 0.0f;
#pragma unroll
    for (int i = 0; i < 8; ++i) tot += reds[i];
    lse[row] = mx + logf(tot);
  }
}

__global__ void k_sub_lse(float* __restrict__ out, const float* __restrict__ lse, size_t n) {
  size_t i = (size_t)blockIdx.x * blockDim.x + threadIdx.x;
  size_t s = (size_t)gridDim.x * blockDim.x;
  for (; i < n; i += s) out[i] -= lse[i / NUM_NODE];
}

// ---------------- launch ----------------

extern "C" void kernel_launch(void* const* d_in, const int* in_sizes, int n_in,
                              void* d_out, int out_size, void* d_ws, size_t ws_size,
                              hipStream_t stream) {
  const int*   iid     = (const int*)d_in[0];
  const int*   dis     = (const int*)d_in[1];
  const int*   pid     = (const int*)d_in[2];
  const int*   tidv    = (const int*)d_in[3];
  const int*   src_int = (const int*)d_in[4];
  const int*   dst_int = (const int*)d_in[5];
  const int*   src_agg = (const int*)d_in[6];
  const int*   dst_agg = (const int*)d_in[7];
  const float* emb     = (const float*)d_in[8];
  const float* pos_emb = (const float*)d_in[9];
  const float* dis1    = (const float*)d_in[10];
  const float* dis2    = (const float*)d_in[11];
  const float* tgt_emb = (const float*)d_in[12];
  const float* w_pi1   = (const float*)d_in[13];
  const float* w_pi2   = (const float*)d_in[14];
  const float* w_q     = (const float*)d_in[15];
  float* out = (float*)d_out;

  float* ws = (float*)d_ws;
  size_t o = 0;
  float* h_v    = ws + o; o += (size_t)N_ITEM * D;
  float* h_acc  = ws + o; o += (size_t)N_ITEM * D;
  float* bnorm  = ws + o; o += (size_t)NUM_NODE * D;
  float* e_buf  = ws + o; o += (size_t)E_INT;
  float* ex_buf = ws + o; o += (size_t)E_INT;
  float* m_buf  = ws + o; o += (size_t)N_ITEM;
  float* denom  = ws + o; o += (size_t)N_ITEM;
  float* sr     = ws + o; o += (size_t)N_TGT * D;
  float* lse    = ws + o; o += (size_t)N_TGT;

  // h_v = l2norm(emb[iid]); bnorm = l2norm(emb)
  k_l2norm<<<N_ITEM, 256, 0, stream>>>(emb, iid, h_v);
  k_l2norm<<<NUM_NODE, 256, 0, stream>>>(emb, nullptr, bnorm);

  // residual accumulator: h = 2*h_v + gat1 + gat2
  k_scale2<<<2048, 256, 0, stream>>>(h_v, h_acc, (size_t)N_ITEM * D);

  // GAT layer 1 (src_int -> dst_int, dis_emb1, w_pi1)
  k_fill<<<160, 256, 0, stream>>>(m_buf, -INFINITY, (size_t)N_ITEM);
  k_fill<<<160, 256, 0, stream>>>(denom, 0.0f, (size_t)N_ITEM);
  k_gat_edge_e<<<E_INT / 8, 256, 0, stream>>>(h_v, dis1, dis, src_int, dst_int, w_pi1,
                                              e_buf, m_buf, E_INT);
  k_gat_expsum<<<E_INT / 256, 256, 0, stream>>>(e_buf, dst_int, m_buf, ex_buf, denom, E_INT);
  k_gat_scatter<<<E_INT, 256, 0, stream>>>(h_v, ex_buf, denom, src_int, dst_int, h_acc);

  // GAT layer 2 (reversed graph: dst_int -> src_int, dis_emb2, w_pi2)
  k_fill<<<160, 256, 0, stream>>>(m_buf, -INFINITY, (size_t)N_ITEM);
  k_fill<<<160, 256, 0, stream>>>(denom, 0.0f, (size_t)N_ITEM);
  k_gat_edge_e<<<E_INT / 8, 256, 0, stream>>>(h_v, dis2, dis, dst_int, src_int, w_pi2,
                                              e_buf, m_buf, E_INT);
  k_gat_expsum<<<E_INT / 256, 256, 0, stream>>>(e_buf, src_int, m_buf, ex_buf, denom, E_INT);
  k_gat_scatter<<<E_INT, 256, 0, stream>>>(h_v, ex_buf, denom, dst_int, src_int, h_acc);

  // PosAggregator -> sr
  k_fill<<<256, 256, 0, stream>>>(sr, 0.0f, (size_t)N_TGT * D);
  k_posagg<<<E_AGG / 16, 128, 0, stream>>>(h_acc, pos_emb, tgt_emb, pid, tidv,
                                           src_agg, dst_agg, w_q, sr);

  // sr = l2norm(sr) in place
  k_l2norm<<<N_TGT, 256, 0, stream>>>(sr, nullptr, sr);

  // out = 12 * sr @ bnorm^T  (fp32 WMMA, TDM-staged A tile)
  dim3 gg(NUM_NODE / 64, N_TGT / 32);
  k_logits<<<gg, 128, 0, stream>>>(sr, bnorm, out);

  // log_softmax in place
  k_rowlse<<<N_TGT, 256, 0, stream>>>(out, lse);
  k_sub_lse<<<4096, 256, 0, stream>>>(out, lse, (size_t)N_TGT * NUM_NODE);
}